// MyConv2d_28913719836970
// MI455X (gfx1250) — compile-verified
//
#include <hip/hip_runtime.h>

typedef __attribute__((ext_vector_type(2))) float v2f;
typedef __attribute__((ext_vector_type(4))) float v4f;
typedef __attribute__((ext_vector_type(8))) float v8f;

namespace {
constexpr int CIN   = 16;
constexpr int COUT  = 16;
constexpr int HW    = 512;            // H == W == 512 (stride 1, pad 1, k3 -> same out size)
constexpr int KDIM  = CIN * 9;        // 144, unfold order (c, kh, kw)
constexpr int NSTEP = KDIM / 4;       // 36 x V_WMMA_F32_16X16X4_F32 per tile
constexpr int TILE_W = 128;           // patches per block-tile = 8 waves x 16
constexpr int SEGW   = 136;           // staged cols: w0-4 .. w0+131 (16B-aligned vecs)
constexpr int COLS4  = SEGW / 4;      // 34 float4 per segment
constexpr int NSEG   = CIN * 3;       // 48 (c, kh) segments staged
constexpr int NBLOCKS = 4096;         // persistent blocks (grid-stride over tiles)
}

// Persistent block = 256 threads = 8 wave32.
//   A (WMMA)  = weights W (16 x 144), preloaded into 72 VGPRs per lane.
//   B (WMMA)  = im2col patches from an LDS-staged x tile; per K-step the two
//               elements sit at compile-time immediate offsets from one of two
//               per-lane base pointers (row stride R: deltas +2 and +(R-1)).
//   D         = (out_channel x patch): each lane owns 8 consecutive output
//               floats -> one 32B fully-coalesced store per tile.
// Output uses the reference's buggy-view layout: out[(b*P + p)*16 + o].
__global__ __launch_bounds__(256) void myconv_wmma_f32(
    const float* __restrict__ x,     // (32, 16, 512, 512)
    const float* __restrict__ Wm,    // (16, 144)
    const float* __restrict__ bias,  // (16,)
    float* __restrict__ out,
    int totalTiles)
{
    __shared__ float sx[NSEG * SEGW];   // 26112 B

    const int tid  = threadIdx.x;
    const int lane = tid & 31;
    const int wave = tid >> 5;
    const int half = lane >> 4;       // K-half per WMMA A/B register layout
    const int m    = lane & 15;       // A row (out channel) == B col (patch idx)
    const int L    = wave * 16 + m;   // lane's patch column within staged tile

    // ---- preload A = W fragments into VGPRs (once per block) ----
    // A[o][k]: lane holds W[o=m][K4+2*half], W[o=m][K4+2*half+1]; 8B aligned.
    v2f Wf[NSTEP];
    {
        const float* wp = Wm + m * KDIM + 2 * half;
        #pragma unroll
        for (int kk = 0; kk < NSTEP; ++kk)
            Wf[kk] = *(const v2f*)(wp + 4 * kk);
    }
    // Bias along M: lane needs bias[8*half + r], r=0..7 -> one 32B load.
    const v8f bv8 = *(const v8f*)(bias + 8 * half);

    // Patch element K lives at sx[(K/3)*SEGW + K%3 + L + 3] (+3: tile starts
    // at col w0-4). Two bases absorb the half-dependent K offset (all kk%3
    // residues verified for row stride R = SEGW).
    const float* pAa = &sx[L + 3 + 2 * half];
    const float* pAb = &sx[L + 3 + (SEGW - 1) * half];

    for (unsigned t = blockIdx.x; t < (unsigned)totalTiles; t += NBLOCKS) {
        // tile -> (b, h, w0): 4 tiles per image row, 512 rows, B images
        const int w0 = (int)(t & 3) * TILE_W;
        const int h  = (int)((t >> 2) & 511);
        const int b  = (int)(t >> 11);
        const float* xb = x + (size_t)b * CIN * HW * HW;

        __syncthreads();   // previous iteration's readers are done with sx

        // ---- stage x tile: 48 segments x 34 float4, zero-padded borders ----
        // Validity is all-or-nothing per vec: cols are 4-aligned, and the
        // boundaries (0, 512) are multiples of 4.
        #pragma unroll
        for (int ss = 0; ss < NSEG / 8; ++ss) {       // 6 segments per wave
            const int s  = wave * (NSEG / 8) + ss;
            const int c  = s / 3;
            const int i  = s - 3 * c;
            const int hh = h + i - 1;
            const bool hok = (unsigned)hh < (unsigned)HW;
            const float* row = xb + (c * HW + (hok ? hh : 0)) * HW;
            #pragma unroll
            for (int it = 0; it < 2; ++it) {
                const int cl4 = lane + it * 32;       // float4 index 0..33
                if (cl4 < COLS4) {
                    const int col = w0 - 4 + 4 * cl4; // first input column
                    v4f v = {0.0f, 0.0f, 0.0f, 0.0f};
                    if (hok && (unsigned)col < (unsigned)HW)
                        v = *(const v4f*)(row + col); // global_load_b128
                    *(v4f*)&sx[s * SEGW + 4 * cl4] = v; // ds_store_b128
                }
            }
        }
        __syncthreads();

        // ---- 36-step K loop: 2x ds_load_b32 + 1x wmma per step ----
        v8f acc = bv8;
        #pragma unroll
        for (int kk = 0; kk < NSTEP; ++kk) {
            const int K4 = kk * 4;
            const int r0 = K4 / 3, j0 = K4 - 3 * r0;
            const int S0 = r0 * SEGW + j0;            // compile-time immediate
            v2f bp;                                   // patch fragment (WMMA B)
            if (kk % 3 == 0)      { bp[0] = pAa[S0]; bp[1] = pAb[S0 + 1]; }
            else if (kk % 3 == 1) { bp[0] = pAb[S0]; bp[1] = pAb[S0 + 1]; }
            else                  { bp[0] = pAb[S0]; bp[1] = pAa[S0 + (SEGW - 2)]; }

            acc = __builtin_amdgcn_wmma_f32_16x16x4_f32(
                false, Wf[kk], false, bp, (short)0, acc, false, false);
        }

        // ---- store: lane owns out channels (8*half..8*half+7) of one patch:
        //      8 consecutive floats, 32B aligned, wave covers 1KB contiguous.
        float* op = out + ((size_t)b * (HW * HW) + (size_t)(h * HW + w0 + L)) * COUT
                        + 8 * half;
        *(v8f*)op = acc;                              // 2x global_store_b128
    }
}

extern "C" void kernel_launch(void* const* d_in, const int* in_sizes, int n_in,
                              void* d_out, int out_size, void* d_ws, size_t ws_size,
                              hipStream_t stream) {
    const float* x    = (const float*)d_in[0];
    const float* Wm   = (const float*)d_in[1];
    const float* bias = (const float*)d_in[2];
    float* out        = (float*)d_out;

    const int B = in_sizes[0] / (CIN * HW * HW);            // 32
    const int totalTiles = B * HW * (HW / TILE_W);          // 65536 block-tiles

    hipLaunchKernelGGL(myconv_wmma_f32, dim3(NBLOCKS), dim3(256), 0, stream,
                       x, Wm, bias, out, totalTiles);
}